// NodeModel_81930796138623
// MI455X (gfx1250) — compile-verified
//
#include <hip/hip_runtime.h>
#include <hip/hip_bf16.h>

#define N_NODES 100000
#define N_EDGES 3200000
#define EDGE_OUT 16
#define NTILES (N_NODES / 16)   // 6250 exactly: no partial tiles

typedef __attribute__((ext_vector_type(2))) float v2f;
typedef __attribute__((ext_vector_type(8))) float v8f;

// ---------------------------------------------------------------------------
// Kernel 1: zero the segment-sum accumulator (workspace is not re-poisoned
// between graph replays, so clear it on every call).
// ---------------------------------------------------------------------------
__global__ void zero_agg_kernel(float* __restrict__ agg, int n) {
  int i = blockIdx.x * blockDim.x + threadIdx.x;
  if (i < n) agg[i] = 0.f;
}

// ---------------------------------------------------------------------------
// Kernel 2: segment_sum(edge_attr, col) via hardware FP32 global atomics.
// Thread t handles (edge = t>>2, quad = t&3): coalesced float4 reads of the
// edge feature rows (512B per wave), scattered with 4x global_atomic_add_f32
// into agg (6.4MB accumulator -> resolves in L2). This is the roofline cost:
// ~235MB of traffic => ~10us at 23.3 TB/s.
// ---------------------------------------------------------------------------
__global__ void scatter_add_kernel(const float* __restrict__ edge_attr,
                                   const int* __restrict__ col,
                                   float* __restrict__ agg) {
  int t = blockIdx.x * blockDim.x + threadIdx.x;
  if (t >= N_EDGES * 4) return;
  int edge = t >> 2;
  int q = t & 3;
  int c = col[edge];
  float4 v = reinterpret_cast<const float4*>(edge_attr)[edge * 4 + q];
  float* dst = agg + (size_t)c * EDGE_OUT + q * 4;
  unsafeAtomicAdd(dst + 0, v.x);
  unsafeAtomicAdd(dst + 1, v.y);
  unsafeAtomicAdd(dst + 2, v.z);
  unsafeAtomicAdd(dst + 3, v.w);
}

// ---------------------------------------------------------------------------
// Kernel 3: per-node MLP + analytic formulas. One wave32 per tile of 16 nodes.
// Layer 1 (h[16x19] @ W1[19x2] + b1) as D = A x B with V_WMMA_F32_16X16X4_F32:
//   A = W1^T (rows 0,1 valid, rest zero), built from uniform registers with
//       compile-time indices (no register-array dynamic indexing).
//   B = h^T K-slices, read from an LDS-staged padded tile (ds_load_b64 with
//       per-lane dynamic address -- native, no cndmask trees).
// After the 5-step K chain, lane n (n<16) holds hidden0=c[0], hidden1=c[1]
// for node n. Only MLP output column 2 survives in the reference.
// ---------------------------------------------------------------------------
__global__ __launch_bounds__(256) void
node_model_kernel(const float* __restrict__ x,
                  const float* __restrict__ agg,
                  const float* __restrict__ W1,
                  const float* __restrict__ b1,
                  const float* __restrict__ W2,
                  const float* __restrict__ b2,
                  float* __restrict__ out) {
  __shared__ float hbuf[8][16][32];           // per-wave h^T tile, 16KB/block

  const int wave = threadIdx.x >> 5;
  const int lane = threadIdx.x & 31;
  const int tile = blockIdx.x * 8 + wave;
  const int half = lane >> 4;                 // 0: K=+0,+1   1: K=+2,+3
  const int n    = lane & 15;                 // node-in-tile == B column == A row
  const bool valid = (tile < NTILES);

  // ---- stage h rows into LDS: row i = node tile*16+i, column = lane -------
  // h row layout: [x0,x1,x2, agg0..agg15, 0...0] padded to 32 floats.
  if (valid) {
#pragma unroll
    for (int i = 0; i < 16; ++i) {
      const int node = tile * 16 + i;
      const float* p = (lane < 3)
          ? (x + (size_t)node * 3 + lane)
          : (agg + (size_t)node * 16 + ((lane < 19) ? (lane - 3) : 12));
      float v = *p;                           // lanes >=19 load a dummy (valid) addr
      if (lane >= 19) v = 0.f;
      hbuf[wave][i][lane] = v;
    }
  } else {
#pragma unroll
    for (int i = 0; i < 16; ++i) hbuf[wave][i][lane] = 0.f;
  }

  // ---- W1 into uniform registers (all indices compile-time) ---------------
  float w[38];
#pragma unroll
  for (int i = 0; i < 38; ++i) w[i] = W1[i];

  __syncthreads();                            // hbuf visible (all waves arrive)

  // ---- WMMA K-chain -------------------------------------------------------
  v8f c = {};
  c[0] = b1[0];                               // C rows 0,1 = b1 (others unused)
  c[1] = b1[1];

#pragma unroll
  for (int s = 0; s < 5; ++s) {
    v2f a, b;
#pragma unroll
    for (int j = 0; j < 2; ++j) {
      const int klo = 4 * s + j;              // compile-time K for half==0
      const int khi = klo + 2;                // compile-time K for half==1
      // A[m][k] = (m<2 && k<19) ? W1[k*2+m] : 0
      float vlo = (n == 0) ? w[klo * 2] : ((n == 1) ? w[klo * 2 + 1] : 0.f);
      float vhi = 0.f;
      if (khi < 19)                           // compile-time predicate (K=19 pad)
        vhi = (n == 0) ? w[khi * 2] : ((n == 1) ? w[khi * 2 + 1] : 0.f);
      a[j] = half ? vhi : vlo;
    }
    // B[k][n] = h[node n][k]; offset even -> single aligned ds_load_b64
    const float2 bb = *(const float2*)&hbuf[wave][n][4 * s + 2 * half];
    b[0] = bb.x;
    b[1] = bb.y;
    c = __builtin_amdgcn_wmma_f32_16x16x4_f32(
        /*neg_a=*/false, a, /*neg_b=*/false, b,
        /*c_mod=*/(short)0, c, /*reuse_a=*/false, /*reuse_b=*/false);
  }

  // ---- scalar tail: lanes 0-15 own valid rows M=0,1 -----------------------
  if (valid && half == 0) {
    const int node = tile * 16 + n;
    float h0 = fmaxf(c[0], 0.f);
    float h1 = fmaxf(c[1], 0.f);
    float mlp2 = h0 * W2[2] + h1 * W2[5] + b2[2];   // only column 2 survives

    float vi = hbuf[wave][n][0];
    float s1 = hbuf[wave][n][1];
    float s2 = hbuf[wave][n][2];

    float n1 = (powf(0.7660379f, s2 / 0.3038425f + s1) +
                powf(0.12117091f, s1) / -0.7256157f) *
                   powf(1.2125463f, vi) +
               0.12262904f;
    float n1_n2 =
        0.7872602f -
        sqrtf(logf(powf(0.1562228f,
                        s2 + (s1 + -3.283101f - vi / 0.79082423f) * 0.31992579f) +
                   1.4462701f));

    out[node * 3 + 0] = n1 + vi;
    out[node * 3 + 1] = (n1_n2 - n1) + s1;
    out[node * 3 + 2] = mlp2 + s2;
  }
}

// ---------------------------------------------------------------------------
extern "C" void kernel_launch(void* const* d_in, const int* in_sizes, int n_in,
                              void* d_out, int out_size, void* d_ws, size_t ws_size,
                              hipStream_t stream) {
  const float* x         = (const float*)d_in[0];
  const float* edge_attr = (const float*)d_in[1];
  // d_in[2] = u (unused by the surviving outputs)
  const float* W1        = (const float*)d_in[3];
  const float* b1        = (const float*)d_in[4];
  const float* W2        = (const float*)d_in[5];
  const float* b2        = (const float*)d_in[6];
  const int*   edge_idx  = (const int*)d_in[7];   // (2, N_EDGES); row 1 = col
  // d_in[8] = batch (unused: single graph)
  float* out = (float*)d_out;
  float* agg = (float*)d_ws;                      // N_NODES*16 floats = 6.4 MB

  const int aggN = N_NODES * EDGE_OUT;
  zero_agg_kernel<<<(aggN + 255) / 256, 256, 0, stream>>>(agg, aggN);

  const int scatterThreads = N_EDGES * 4;
  scatter_add_kernel<<<(scatterThreads + 255) / 256, 256, 0, stream>>>(
      edge_attr, edge_idx + N_EDGES, agg);

  const int blocks = (NTILES + 7) / 8;            // 8 wave32 per 256-thread block
  node_model_kernel<<<blocks, 256, 0, stream>>>(x, agg, W1, b1, W2, b2, out);
}